// Gconv_1382979469319
// MI455X (gfx1250) — compile-verified
//
#include <hip/hip_runtime.h>

// ---------------------------------------------------------------------------
// Types for CDNA5 WMMA f32 16x16x4:  A = 16x4 f32 (2 VGPRs), B = 4x16 f32
// (2 VGPRs), C/D = 16x16 f32 (8 VGPRs).
// ---------------------------------------------------------------------------
typedef __attribute__((ext_vector_type(2))) float v2f;
typedef __attribute__((ext_vector_type(8))) float v8f;

#define IN_DIM   2048   // NUM_PT * 8
#define OUT_DIM  4096   // NUM_PT * UP_FTR * OUT_F
#define BATCH    64
#define NPTS     512    // NUM_PT * UP_FTR
#define CH       8      // OUT_F
#define KNN      16

// ---------------------------------------------------------------------------
// Kernel 1: feat = x @ W_mlp + b_mlp   (64 x 2048) @ (2048 x 4096)
//
// Wave tile: 16(M) x 64(N)  -> 4 v8f accumulators.
// Grid: 4 M-tiles * 64 N-groups = 256 waves = 32 blocks of 8 waves.
// K loop steps by 4 (one V_WMMA_F32_16X16X4_F32 per N-subtile per step).
//
// A layout (ISA 7.12.2, 32-bit A 16x4): lane l -> row m = l&15,
//   VGPR0 = K (l>>4)*2, VGPR1 = K (l>>4)*2 + 1.
// B layout (4x16, rows striped across lanes, VGPR r holds rows r and r+2):
//   lane l -> col n = l&15, VGPR0 = K-row (l>>4)*2, VGPR1 = K-row (l>>4)*2+1.
// C/D layout: VGPR r -> row r (lanes 0-15) / row r+8 (lanes 16-31), col l&15.
// ---------------------------------------------------------------------------
__global__ __launch_bounds__(256) void mlp_gemm_wmma_f32(
    const float* __restrict__ x,      // (64, 2048) row-major
    const float* __restrict__ W,      // (2048, 4096) row-major
    const float* __restrict__ bias,   // (4096,)
    float* __restrict__ feat)         // (64, 4096) row-major
{
    const int lane  = threadIdx.x & 31;
    const int wave  = threadIdx.x >> 5;
    const int gw    = blockIdx.x * 8 + wave;   // 0..255
    const int mtile = gw >> 6;                 // 0..3
    const int ngrp  = gw & 63;                 // 0..63
    const int mbase = mtile * 16;
    const int nbase = ngrp * 64;

    const int half = lane >> 4;                // 0 or 1
    const int l16  = lane & 15;
    const int kb   = half * 2;                 // per-lane K sub-offset

    const float* arow = x + (mbase + l16) * IN_DIM + kb;
    const float* bcol = W + nbase + l16;       // + (k+kb)*OUT_DIM + t*16

    v8f acc0 = {}, acc1 = {}, acc2 = {}, acc3 = {};

    #pragma unroll 4
    for (int k = 0; k < IN_DIM; k += 4) {
        // Keep the W stream flowing: prefetch 32 K-rows ahead (L2-resident
        // working set, 192MB L2 -> speculative RT prefetch is cheap).
        __builtin_prefetch(bcol + (k + 32) * OUT_DIM, 0, 1);

        v2f a = *(const v2f*)(arow + k);       // global_load_b64

        const float* brow0 = bcol + (k + kb) * OUT_DIM;
        const float* brow1 = brow0 + OUT_DIM;

        v2f b0, b1, b2, b3;
        b0.x = brow0[0];   b0.y = brow1[0];
        b1.x = brow0[16];  b1.y = brow1[16];
        b2.x = brow0[32];  b2.y = brow1[32];
        b3.x = brow0[48];  b3.y = brow1[48];

        acc0 = __builtin_amdgcn_wmma_f32_16x16x4_f32(false, a, false, b0,
                                                     (short)0, acc0, false, false);
        acc1 = __builtin_amdgcn_wmma_f32_16x16x4_f32(false, a, false, b1,
                                                     (short)0, acc1, false, false);
        acc2 = __builtin_amdgcn_wmma_f32_16x16x4_f32(false, a, false, b2,
                                                     (short)0, acc2, false, false);
        acc3 = __builtin_amdgcn_wmma_f32_16x16x4_f32(false, a, false, b3,
                                                     (short)0, acc3, false, false);
    }

    // Epilogue: bias add + store per documented C/D layout.
    const float bs0 = bias[nbase + l16];
    const float bs1 = bias[nbase + 16 + l16];
    const float bs2 = bias[nbase + 32 + l16];
    const float bs3 = bias[nbase + 48 + l16];

    #pragma unroll
    for (int r = 0; r < 8; ++r) {
        const int row = mbase + r + half * 8;
        float* o = feat + row * OUT_DIM + nbase + l16;
        o[0]  = acc0[r] + bs0;
        o[16] = acc1[r] + bs1;
        o[32] = acc2[r] + bs2;
        o[48] = acc3[r] + bs3;
    }
}

// ---------------------------------------------------------------------------
// Kernel 2: per batch b:
//   fs   = feat[b] as (C=8, N=512) in LDS
//   ss   = support = feat[b]^T @ W_gcn  as (N=512, O=8) in LDS
//   for each point n: top-16 nearest (squared L2 over 8 channels; sqrt is
//   monotone so ranking by d^2 matches the reference's top_k on -dist),
//   out[b, o, n] = mean(support over 16 neighbors) + b_gcn[o]
// One 512-thread block per batch; 32KB LDS (<< 320KB/WGP).
// ---------------------------------------------------------------------------
__global__ __launch_bounds__(512) void knn_gconv(
    const float* __restrict__ feat,   // (64, 4096) = (64, 8, 512)
    const float* __restrict__ Wg,     // (8, 8) row-major (C, O)
    const float* __restrict__ bg,     // (8, 1)
    float* __restrict__ out)          // (64, 4096) = (64, 8, 512)
{
    __shared__ float fs[CH][NPTS];    // 16 KB
    __shared__ float ss[NPTS][CH];    // 16 KB

    const int b = blockIdx.x;
    const int n = threadIdx.x;        // 0..511
    const float* fb = feat + b * (CH * NPTS);

    // Stage feat[b] into LDS: 4096 floats, 512 threads -> 8 each.
    #pragma unroll
    for (int i = 0; i < CH; ++i)
        fs[i][n] = fb[i * NPTS + n];
    __syncthreads();

    // This thread's own channel vector.
    float fn[CH];
    #pragma unroll
    for (int c = 0; c < CH; ++c) fn[c] = fs[c][n];

    // support[n][o] = sum_c fn[c] * Wg[c][o]
    #pragma unroll
    for (int o = 0; o < CH; ++o) {
        float s = 0.f;
        #pragma unroll
        for (int c = 0; c < CH; ++c) s = fmaf(fn[c], Wg[c * CH + o], s);
        ss[n][o] = s;
    }
    __syncthreads();

    // Sorted-16 smallest (ascending) kept fully in registers: all indices are
    // compile-time after unrolling, so this lowers to v_cndmask chains.
    float best[KNN];
    int   bidx[KNN];
    #pragma unroll
    for (int j = 0; j < KNN; ++j) { best[j] = 3.4e38f; bidx[j] = 0; }

    for (int m = 0; m < NPTS; ++m) {
        // d^2(n, m); fs[c][m] is uniform across the wave -> LDS broadcast.
        float d = 0.f;
        #pragma unroll
        for (int c = 0; c < CH; ++c) {
            float t = fn[c] - fs[c][m];
            d = fmaf(t, t, d);
        }
        int id = m;
        #pragma unroll
        for (int j = 0; j < KNN; ++j) {
            const bool  sw = d < best[j];
            const float td = sw ? best[j] : d;
            const int   ti = sw ? bidx[j] : id;
            best[j] = sw ? d  : best[j];
            bidx[j] = sw ? id : bidx[j];
            d = td; id = ti;
        }
    }

    // Average support over the 16 neighbors.
    float acc[CH];
    #pragma unroll
    for (int o = 0; o < CH; ++o) acc[o] = 0.f;
    #pragma unroll
    for (int j = 0; j < KNN; ++j) {
        const float* sp = ss[bidx[j]];
        #pragma unroll
        for (int o = 0; o < CH; ++o) acc[o] += sp[o];
    }

    float* ob = out + b * (CH * NPTS);
    #pragma unroll
    for (int o = 0; o < CH; ++o)
        ob[o * NPTS + n] = acc[o] * (1.0f / KNN) + bg[o];
}

// ---------------------------------------------------------------------------
// Launch
// ---------------------------------------------------------------------------
extern "C" void kernel_launch(void* const* d_in, const int* in_sizes, int n_in,
                              void* d_out, int out_size, void* d_ws, size_t ws_size,
                              hipStream_t stream) {
    (void)in_sizes; (void)n_in; (void)out_size; (void)ws_size;

    const float* x     = (const float*)d_in[0];   // (64, 2048)
    const float* W_mlp = (const float*)d_in[1];   // (2048, 4096)
    const float* b_mlp = (const float*)d_in[2];   // (4096,)
    const float* W_gcn = (const float*)d_in[3];   // (8, 8)
    const float* b_gcn = (const float*)d_in[4];   // (8, 1)
    float*       outp  = (float*)d_out;           // (64, 4096)
    float*       feat  = (float*)d_ws;            // (64, 4096) scratch, 1 MB

    mlp_gemm_wmma_f32<<<dim3(32), dim3(256), 0, stream>>>(x, W_mlp, b_mlp, feat);
    knn_gconv<<<dim3(64), dim3(512), 0, stream>>>(feat, W_gcn, b_gcn, outp);
}